// DGFEM_2843268350233
// MI455X (gfx1250) — compile-verified
//
#include <hip/hip_runtime.h>
#include <hip/hip_bf16.h>
#include <stdint.h>

// ---------------------------------------------------------------------------
// Fused DGFEM-style block for MI455X (gfx1250, wave32).
// Memory-bound (AI ~16 FLOP/B) -> single-pass fusion:
//   * x0 tile streamed HBM->LDS with GLOBAL_LOAD_ASYNC_TO_LDS_B32 (ASYNCcnt),
//     issued at kernel start to overlap with the tiny MLPs.
//   * channel-mixing 64x64 1x1 conv via fp32 WMMA (V_WMMA_F32_16X16X4_F32).
// ---------------------------------------------------------------------------

#define CC 64
#define TS 16                 // pixel tile 16x16
#define HALO 18               // TS + 2 (3x3, pad 1)
#define XS_CH (HALO * HALO)   // 324 floats per channel
#define YS_STRIDE 272         // 256 pixels + 16 pad  (2*272 % 64 == 32 -> lane halves hit disjoint banks)
#define WC_STRIDE 68          // 64 + 4 pad           (conflict-free A-fragment b64 loads)

typedef float v2f __attribute__((ext_vector_type(2)));
typedef float v8f __attribute__((ext_vector_type(8)));

// smem layout (floats):
//   v_s[64] hca[8] hk[64] att[64] bias[64] kern[576] wc[64*68] xs[64*324] ys[64*272]
#define SM_V     0
#define SM_HCA   (SM_V + 64)
#define SM_HK    (SM_HCA + 8)
#define SM_ATT   (SM_HK + 64)
#define SM_BIAS  (SM_ATT + 64)
#define SM_KERN  (SM_BIAS + 64)
#define SM_WC    (SM_KERN + 576)
#define SM_XS    (SM_WC + CC * WC_STRIDE)
#define SM_YS    (SM_XS + CC * XS_CH)
#define SM_TOTAL (SM_YS + CC * YS_STRIDE)   // 43336 floats = 173344 bytes (< 320KB WGP LDS)

__device__ __forceinline__ float lrelu01(float a) {
    // leaky_relu slope 0.1:  max(a, 0.1*a) is exact for both signs
    return fmaxf(a, 0.1f * a);
}

__launch_bounds__(256, 1)
__global__ void dgfem_fused(const float* __restrict__ x0,
                            const float* __restrict__ v,
                            const float* __restrict__ ca_w1,   // [8,64]
                            const float* __restrict__ ca_w2,   // [64,8]
                            const float* __restrict__ k_w1,    // [64,64]
                            const float* __restrict__ k_w2,    // [576,64]
                            const float* __restrict__ conv_w,  // [64,64]
                            const float* __restrict__ conv_b,  // [64]
                            float* __restrict__ out,
                            int H, int W)
{
    extern __shared__ float sm[];
    float* v_s    = sm + SM_V;
    float* hca    = sm + SM_HCA;
    float* hk     = sm + SM_HK;
    float* att    = sm + SM_ATT;
    float* bias_s = sm + SM_BIAS;
    float* kern_s = sm + SM_KERN;
    float* wc_s   = sm + SM_WC;
    float* xs     = sm + SM_XS;
    float* ys     = sm + SM_YS;

    const int t   = threadIdx.x;
    const int b   = blockIdx.z;
    const int ty0 = blockIdx.y * TS;
    const int tx0 = blockIdx.x * TS;

    // ---- stage A (first!): async HBM -> LDS stream of the x0 tile ---------
    // GLOBAL_LOAD_ASYNC_TO_LDS_B32: no VGPR staging, tracked on ASYNCcnt, so
    // the DRAM latency overlaps the MLP stages below. OOB halo lanes write 0
    // through the normal DS path (disjoint addresses -> no race).
    for (int i = t; i < CC * XS_CH; i += 256) {
        const int c   = i / XS_CH;
        const int rem = i - c * XS_CH;
        const int yy  = rem / HALO;
        const int xx  = rem - yy * HALO;
        const int gy  = ty0 + yy - 1;
        const int gx  = tx0 + xx - 1;
        if (gy >= 0 && gy < H && gx >= 0 && gx < W) {
            const uint64_t gaddr =
                (uint64_t)(uintptr_t)(x0 + (((size_t)(b * CC + c) * H + gy) * W + gx));
            const uint32_t laddr = (uint32_t)(uintptr_t)(xs + i);  // low 32b of flat = LDS offset
            asm volatile("global_load_async_to_lds_b32 %0, %1, off"
                         :: "v"(laddr), "v"(gaddr) : "memory");
        } else {
            xs[i] = 0.f;
        }
    }

    // ---- stage 0: embedding + bias + conv_w (padded) into LDS -------------
    if (t < CC) { v_s[t] = v[b * CC + t]; bias_s[t] = conv_b[t]; }
    for (int i = t; i < CC * CC; i += 256) {
        int o = i >> 6, c = i & 63;
        wc_s[o * WC_STRIDE + c] = conv_w[i];
    }
    __syncthreads();

    // ---- stage 1: MLP hidden layers (overlaps the async tile stream) ------
    if (t < 8) {                       // CA hidden (8)
        float a = 0.f;
        #pragma unroll 8
        for (int c = 0; c < CC; ++c) a += v_s[c] * ca_w1[t * CC + c];
        hca[t] = lrelu01(a);
    }
    if (t >= 64 && t < 128) {          // kernel-MLP hidden (64)
        const int j = t - 64;
        float a = 0.f;
        #pragma unroll 8
        for (int c = 0; c < CC; ++c) a += v_s[c] * k_w1[j * CC + c];
        hk[j] = lrelu01(a);
    }
    __syncthreads();

    // ---- stage 2: channel attention gate ----------------------------------
    if (t < CC) {
        float a = 0.f;
        #pragma unroll
        for (int r = 0; r < 8; ++r) a += hca[r] * ca_w2[t * 8 + r];
        att[t] = 1.f / (1.f + __expf(-a));
    }
    __syncthreads();

    // ---- stage 3: per-(b,c) 3x3 kernels, att folded in (depthwise conv is
    //               linear per channel: conv(x*att_c,k_c) == att_c*conv(x,k_c))
    for (int q = t; q < CC * 9; q += 256) {
        float a = 0.f;
        #pragma unroll 8
        for (int j = 0; j < CC; ++j) a += hk[j] * k_w2[q * CC + j];
        kern_s[q] = a * att[q / 9];
    }

    // ---- drain the async tile stream, then make it workgroup-visible ------
    asm volatile("s_wait_asynccnt 0" ::: "memory");
    __syncthreads();   // supplies dscnt wait (zero-fill stores) + barrier

    // ---- stage 5: depthwise 3x3 + lrelu -> ys[c][pixel] -------------------
    {
        const int px = t & 15, py = t >> 4;   // this thread's pixel
        for (int c = 0; c < CC; ++c) {
            const float* xc = xs + c * XS_CH + py * HALO + px;
            const float* kc = kern_s + c * 9;
            float a = xc[0] * kc[0] + xc[1] * kc[1] + xc[2] * kc[2]
                    + xc[HALO]     * kc[3] + xc[HALO + 1]     * kc[4] + xc[HALO + 2]     * kc[5]
                    + xc[2 * HALO] * kc[6] + xc[2 * HALO + 1] * kc[7] + xc[2 * HALO + 2] * kc[8];
            ys[c * YS_STRIDE + t] = lrelu01(a);
        }
    }
    __syncthreads();

    // ---- stage 6: 1x1 conv as fp32 WMMA GEMM ------------------------------
    // D[o,p] = sum_c conv_w[o,c] * y[c,p] ; A = conv_w (16x4), B = y (4x16).
    // ISA layouts: A lane=M, VGPR0/1 = K{0,1} (lanes 16-31 -> K{2,3});
    //              B lane=N, same K offset per lane-half; D VGPR r -> M = r + 8*half.
    const int lane = t & 31;
    const int wave = t >> 5;
    const int n    = lane & 15;
    const int half = lane >> 4;

    for (int pg = wave; pg < 16; pg += 8) {          // uniform per wave: EXEC all-1s
        const int p0 = pg * 16;
        v8f acc0 = {}, acc1 = {}, acc2 = {}, acc3 = {};

        #pragma unroll 4
        for (int kk = 0; kk < 16; ++kk) {
            const int ch = kk * 4 + 2 * half;        // this lane-half's K rows
            v2f bfrag;
            bfrag.x = ys[ch * YS_STRIDE + p0 + n];
            bfrag.y = ys[(ch + 1) * YS_STRIDE + p0 + n];
            // A fragments: consecutive K pair -> 8B-aligned b64 LDS loads
            const v2f a0 = *(const v2f*)(wc_s + (0 * 16 + n) * WC_STRIDE + ch);
            const v2f a1 = *(const v2f*)(wc_s + (1 * 16 + n) * WC_STRIDE + ch);
            const v2f a2 = *(const v2f*)(wc_s + (2 * 16 + n) * WC_STRIDE + ch);
            const v2f a3 = *(const v2f*)(wc_s + (3 * 16 + n) * WC_STRIDE + ch);
            acc0 = __builtin_amdgcn_wmma_f32_16x16x4_f32(false, a0, false, bfrag, (short)0, acc0, false, false);
            acc1 = __builtin_amdgcn_wmma_f32_16x16x4_f32(false, a1, false, bfrag, (short)0, acc1, false, false);
            acc2 = __builtin_amdgcn_wmma_f32_16x16x4_f32(false, a2, false, bfrag, (short)0, acc2, false, false);
            acc3 = __builtin_amdgcn_wmma_f32_16x16x4_f32(false, a3, false, bfrag, (short)0, acc3, false, false);
        }

        // epilogue: bias + store (16 lanes -> 64B contiguous per output row)
        const int p  = p0 + n;
        const int py = p >> 4, px = p & 15;
        const int gy = ty0 + py, gx = tx0 + px;
        float* op = out + ((size_t)(b * CC) * H + gy) * W + gx;
        #pragma unroll
        for (int r = 0; r < 8; ++r) {
            const int o = r + 8 * half;
            op[(size_t)(o)      * H * W] = acc0[r] + bias_s[o];
            op[(size_t)(o + 16) * H * W] = acc1[r] + bias_s[o + 16];
            op[(size_t)(o + 32) * H * W] = acc2[r] + bias_s[o + 32];
            op[(size_t)(o + 48) * H * W] = acc3[r] + bias_s[o + 48];
        }
    }
}

extern "C" void kernel_launch(void* const* d_in, const int* in_sizes, int n_in,
                              void* d_out, int out_size, void* d_ws, size_t ws_size,
                              hipStream_t stream) {
    const float* x0     = (const float*)d_in[0];
    const float* v      = (const float*)d_in[1];
    const float* ca_w1  = (const float*)d_in[2];
    const float* ca_w2  = (const float*)d_in[3];
    const float* k_w1   = (const float*)d_in[4];
    const float* k_w2   = (const float*)d_in[5];
    const float* conv_w = (const float*)d_in[6];
    const float* conv_b = (const float*)d_in[7];
    float* out = (float*)d_out;

    const int B = in_sizes[1] / CC;   // 32
    const int H = 128, W = 128;

    const size_t smem = (size_t)SM_TOTAL * sizeof(float);   // ~173 KB (< 320 KB WGP LDS)
    (void)hipFuncSetAttribute(reinterpret_cast<const void*>(dgfem_fused),
                              hipFuncAttributeMaxDynamicSharedMemorySize, (int)smem);

    dim3 grid(W / TS, H / TS, B);
    dgfem_fused<<<grid, 256, smem, stream>>>(x0, v, ca_w1, ca_w2, k_w1, k_w2,
                                             conv_w, conv_b, out, H, W);
    (void)d_ws; (void)ws_size; (void)n_in; (void)out_size;
}